// MACE_62345745268786
// MI455X (gfx1250) — compile-verified
//
#include <hip/hip_runtime.h>

typedef float v2f __attribute__((ext_vector_type(2)));
typedef float v8f __attribute__((ext_vector_type(8)));

// problem sizes (match reference)
constexpr int NN   = 2048;
constexpr int C    = 128;
constexpr int L    = 16;
constexpr int E    = 10;
constexpr int K3_0 = 23, K2_0 = 4;
constexpr int K3_1 = 33, K2_1 = 6;

// combined B-matrix columns: [A1 (3*33) | A0 (23) | U2_1 (3*6) | U2_0 (4)] = 144
constexpr int COL_A1 = 0;
constexpr int COL_A0 = 3 * K3_1;            // 99
constexpr int COL_21 = COL_A0 + K3_0;       // 122
constexpr int COL_20 = COL_21 + 3 * K2_1;   // 140
constexpr int NCOL   = COL_20 + K2_0;       // 144
constexpr int NTILE  = NCOL / 16;           // 9

// B stored row-pair interleaved: sB2[pair][col][2], pair = k>>1 (128 pairs)
// pair stride 290 dwords: 290 % 64 == 34 -> conflict-free pair-writes,
// near-conflict-free b64 fragment reads (2-bank overlap between wave halves)
constexpr int S2   = 2 * NCOL + 2;          // 290
constexpr int PSTR = 149;                   // P stride, odd
constexpr int XSTR = L + 1;                 // 17

// per-node weight rows in sW: [wm1 33][wm0 23][w21 6][w20 4][w11 1][w10 1]
constexpr int WR_M1 = 0;
constexpr int WR_M0 = 33;
constexpr int WR_21 = 56;
constexpr int WR_20 = 62;
constexpr int WR_11 = 66;
constexpr int WR_10 = 67;
constexpr int WROWS = 68;

// dynamic LDS layout (floats)
constexpr int OFF_X  = 0;                   // 128*17 = 2176
constexpr int OFF_XS = OFF_X + C * XSTR;    // 16
constexpr int OFF_W  = OFF_XS + L;          // 68*128 = 8704
constexpr int OFF_U1 = OFF_W + WROWS * C;   // 64 : [0..15]=U1_0, [16+w*16+x]=U1_1
constexpr int OFF_B2 = OFF_U1 + 64;         // 128*290 = 37120
constexpr int OFF_P  = OFF_B2 + 128 * S2;   // 128*149 = 19072
constexpr int SMEM_FLOATS = OFF_P + C * PSTR;   // 67152 floats = 268608 B (<320KB/WGP)

// build two adjacent rows (2p, 2p+1) of the A1 block, store as b64 pairs
__device__ __forceinline__ void build_a1_pair(const float* __restrict__ U3_1,
                                              const float* __restrict__ sXS,
                                              float* __restrict__ sB2, int w, int p)
{
    const float* u0 = U3_1 + (size_t)(w * 256 + 2 * p) * (L * K3_1);
    const float* u1 = u0 + L * K3_1;
    float a0[K3_1], a1[K3_1];
    #pragma unroll
    for (int k = 0; k < K3_1; ++k) { a0[k] = 0.f; a1[k] = 0.f; }
    for (int i = 0; i < L; ++i) {
        const float xi = sXS[i];
        #pragma unroll
        for (int k = 0; k < K3_1; ++k) {
            a0[k] = fmaf(u0[i * K3_1 + k], xi, a0[k]);
            a1[k] = fmaf(u1[i * K3_1 + k], xi, a1[k]);
        }
    }
    float* bp = sB2 + p * S2 + (COL_A1 + w * K3_1) * 2;
    #pragma unroll
    for (int k = 0; k < K3_1; ++k)
        *(v2f*)(bp + 2 * k) = (v2f){a0[k], a1[k]};
}

__device__ __forceinline__ void build_a0_pair(const float* __restrict__ U3_0,
                                              const float* __restrict__ sXS,
                                              float* __restrict__ sB2, int p)
{
    const float* u0 = U3_0 + (size_t)(2 * p) * (L * K3_0);
    const float* u1 = u0 + L * K3_0;
    float a0[K3_0], a1[K3_0];
    #pragma unroll
    for (int k = 0; k < K3_0; ++k) { a0[k] = 0.f; a1[k] = 0.f; }
    for (int i = 0; i < L; ++i) {
        const float xi = sXS[i];
        #pragma unroll
        for (int k = 0; k < K3_0; ++k) {
            a0[k] = fmaf(u0[i * K3_0 + k], xi, a0[k]);
            a1[k] = fmaf(u1[i * K3_0 + k], xi, a1[k]);
        }
    }
    float* bp = sB2 + p * S2 + COL_A0 * 2;
    #pragma unroll
    for (int k = 0; k < K3_0; ++k)
        *(v2f*)(bp + 2 * k) = (v2f){a0[k], a1[k]};
}

__global__ __launch_bounds__(256) void mace_node_kernel(
    const float* __restrict__ x,    const float* __restrict__ y,
    const float* __restrict__ U3_0, const float* __restrict__ U2_0, const float* __restrict__ U1_0,
    const float* __restrict__ U3_1, const float* __restrict__ U2_1, const float* __restrict__ U1_1,
    const float* __restrict__ Wm0,  const float* __restrict__ W20,  const float* __restrict__ W10,
    const float* __restrict__ Wm1,  const float* __restrict__ W21,  const float* __restrict__ W11,
    float* __restrict__ out)
{
    extern __shared__ float sm[];
    float* sX  = sm + OFF_X;
    float* sXS = sm + OFF_XS;
    float* sW  = sm + OFF_W;
    float* sU1 = sm + OFF_U1;
    float* sB2 = sm + OFF_B2;
    float* sP  = sm + OFF_P;

    const int tid = threadIdx.x;
    const int b   = blockIdx.x;

    // ---- phase 0: load x[b] tile [C][L] (coalesced) ----
    for (int idx = tid; idx < C * L; idx += 256)
        sX[(idx >> 4) * XSTR + (idx & 15)] = x[(size_t)b * (C * L) + idx];

    // ---- phase 1: per-node weights  sW[wr][c] = sum_e y[b,e] * W[e,k,c] ----
    float yv[E];
    #pragma unroll
    for (int e = 0; e < E; ++e) yv[e] = y[b * E + e];
    for (int idx = tid; idx < WROWS * C; idx += 256) {
        const int wr = idx >> 7, c = idx & 127;
        const float* src; int stride;
        if (wr < WR_M0)       { src = Wm1 + (wr)         * C + c; stride = K3_1 * C; }
        else if (wr < WR_21)  { src = Wm0 + (wr - WR_M0) * C + c; stride = K3_0 * C; }
        else if (wr < WR_20)  { src = W21 + (wr - WR_21) * C + c; stride = K2_1 * C; }
        else if (wr < WR_11)  { src = W20 + (wr - WR_20) * C + c; stride = K2_0 * C; }
        else if (wr == WR_11) { src = W11 + c;                    stride = C;        }
        else                  { src = W10 + c;                    stride = C;        }
        float acc = 0.f;
        #pragma unroll
        for (int e = 0; e < E; ++e) acc = fmaf(yv[e], src[e * stride], acc);
        sW[idx] = acc;
    }
    if (tid < 16)               sU1[tid] = U1_0[tid];        // [L,1]
    else if (tid < 64)          sU1[tid] = U1_1[tid - 16];   // [3,L,1] flat w*16+x
    __syncthreads();

    // ---- channel-sum xs[i] = sum_c x[c,i] ----
    if (tid < L) {
        float s = 0.f;
        for (int c = 0; c < C; ++c) s += sX[c * XSTR + tid];
        sXS[tid] = s;
    }
    __syncthreads();

    // ---- phase 2: build B2[128 pairs][144 cols][2] in LDS ----
    // A1 pair-tasks: 384 total; waves 4-7 take two, waves 0-3 take one + A0 + U2.
    {
        const int t0 = tid;                 // task 0..255
        build_a1_pair(U3_1, sXS, sB2, t0 >> 7, t0 & 127);
        if (tid >= 128) {
            const int t1 = tid + 128;       // tasks 256..383
            build_a1_pair(U3_1, sXS, sB2, t1 >> 7, t1 & 127);
        } else {
            const int p = tid;              // pair 0..127
            build_a0_pair(U3_0, sXS, sB2, p);
            // constant U2 columns, b64 pair stores
            float* bp = sB2 + p * S2;
            #pragma unroll
            for (int m = 0; m < 3 * K2_1; ++m) {
                const int w = m / K2_1, k = m % K2_1;
                const int r0 = (w * 256 + 2 * p) * K2_1 + k;
                *(v2f*)(bp + (COL_21 + m) * 2) = (v2f){U2_1[r0], U2_1[r0 + K2_1]};
            }
            #pragma unroll
            for (int m = 0; m < K2_0; ++m)
                *(v2f*)(bp + (COL_20 + m) * 2) =
                    (v2f){U2_0[(2 * p) * K2_0 + m], U2_0[(2 * p + 1) * K2_0 + m]};
        }
    }
    __syncthreads();

    // ---- phase 3: P[128x144] = X2[128x256] * B[256x144] via V_WMMA_F32_16X16X4_F32 ----
    const int wv   = tid >> 5;           // wave id = M tile (16 channels)
    const int lane = tid & 31;
    const int nl   = lane & 15;
    const int hi   = lane >> 4;          // lanes 0-15: K=k0,k0+1 ; lanes 16-31: k0+2,k0+3
    const int crow = wv * 16 + nl;

    float xr[L];
    #pragma unroll
    for (int j = 0; j < L; ++j) xr[j] = sX[crow * XSTR + j];

    // hoisted hi-dependent picks: value-selects only (keeps xr in registers, 8 cndmasks total)
    float sel0[4], sel1[4];
    #pragma unroll
    for (int j = 0; j < 4; ++j) {
        const float e0 = xr[4 * j],     e1 = xr[4 * j + 1];
        const float e2 = xr[4 * j + 2], e3 = xr[4 * j + 3];
        sel0[j] = hi ? e2 : e0;
        sel1[j] = hi ? e3 : e1;
    }

    v8f acc[NTILE];
    #pragma unroll
    for (int t = 0; t < NTILE; ++t)
        acc[t] = (v8f){0.f, 0.f, 0.f, 0.f, 0.f, 0.f, 0.f, 0.f};

    // per-lane base: pair = 2kk+hi, col base = nl
    const float* Bl = sB2 + hi * S2 + nl * 2;
    #pragma unroll
    for (int kk = 0; kk < 64; ++kk) {          // K = 256, 4 per WMMA
        const float xp = xr[kk >> 2];          // compile-time indices under full unroll
        v2f a;                                  // A frag: X2[crow, k0..k0+1]
        a.x = xp * sel0[kk & 3];
        a.y = xp * sel1[kk & 3];
        const float* Br = Bl + (2 * kk) * S2;
        #pragma unroll
        for (int t = 0; t < NTILE; ++t) {
            const v2f bf = *(const v2f*)(Br + t * 32);   // rows {k0,k0+1}: one ds_load_b64
            acc[t] = __builtin_amdgcn_wmma_f32_16x16x4_f32(
                false, a, false, bf, (short)0, acc[t], false, false);
        }
    }

    // C/D layout: VGPR r -> M = r (lanes 0-15) / r+8 (lanes 16-31), N = nl
    #pragma unroll
    for (int t = 0; t < NTILE; ++t) {
        #pragma unroll
        for (int r = 0; r < 8; ++r)
            sP[(wv * 16 + r + hi * 8) * PSTR + t * 16 + nl] = acc[t][r];
    }
    __syncthreads();

    // ---- phase 4: epilogue — diagonal contraction with per-node weights + U1 terms ----
    for (int idx = tid; idx < 4 * C; idx += 256) {     // wave-uniform g
        const int c = idx & 127;
        const int g = idx >> 7;                        // 0: out0, 1..3: out1 component w=g-1
        const float* P = sP + c * PSTR;
        float s = 0.f;
        if (g == 0) {
            #pragma unroll
            for (int k = 0; k < K3_0; ++k) s = fmaf(P[COL_A0 + k], sW[(WR_M0 + k) * C + c], s);
            #pragma unroll
            for (int k = 0; k < K2_0; ++k) s = fmaf(P[COL_20 + k], sW[(WR_20 + k) * C + c], s);
            float d = 0.f;
            #pragma unroll
            for (int j = 0; j < L; ++j) d = fmaf(sU1[j], sX[c * XSTR + j], d);
            s = fmaf(sW[WR_10 * C + c], d, s);
            out[(size_t)b * (4 * C) + c] = s;
        } else {
            const int w = g - 1;
            #pragma unroll
            for (int k = 0; k < K3_1; ++k) s = fmaf(P[COL_A1 + w * K3_1 + k], sW[(WR_M1 + k) * C + c], s);
            #pragma unroll
            for (int k = 0; k < K2_1; ++k) s = fmaf(P[COL_21 + w * K2_1 + k], sW[(WR_21 + k) * C + c], s);
            float d = 0.f;
            #pragma unroll
            for (int j = 0; j < L; ++j) d = fmaf(sU1[16 + w * L + j], sX[c * XSTR + j], d);
            s = fmaf(sW[WR_11 * C + c], d, s);
            out[(size_t)b * (4 * C) + C + c * 3 + w] = s;   // o1 reshape: [N,C,3] -> c*3+w
        }
    }
}

extern "C" void kernel_launch(void* const* d_in, const int* in_sizes, int n_in,
                              void* d_out, int out_size, void* d_ws, size_t ws_size,
                              hipStream_t stream) {
    (void)in_sizes; (void)n_in; (void)out_size; (void)d_ws; (void)ws_size;
    // setup_inputs() dict order:
    const float* x    = (const float*)d_in[0];
    const float* y    = (const float*)d_in[1];
    const float* U3_0 = (const float*)d_in[2];
    const float* U2_0 = (const float*)d_in[3];
    const float* U1_0 = (const float*)d_in[4];
    const float* U3_1 = (const float*)d_in[5];
    const float* U2_1 = (const float*)d_in[6];
    const float* U1_1 = (const float*)d_in[7];
    const float* Wm0  = (const float*)d_in[8];
    const float* W20  = (const float*)d_in[9];
    const float* W10  = (const float*)d_in[10];
    const float* Wm1  = (const float*)d_in[11];
    const float* W21  = (const float*)d_in[12];
    const float* W11  = (const float*)d_in[13];

    constexpr size_t smem_bytes = (size_t)SMEM_FLOATS * sizeof(float);
    mace_node_kernel<<<dim3(NN), dim3(256), smem_bytes, stream>>>(
        x, y, U3_0, U2_0, U1_0, U3_1, U2_1, U1_1,
        Wm0, W20, W10, Wm1, W21, W11, (float*)d_out);
}